// TripletLoss_49194555408760
// MI455X (gfx1250) — compile-verified
//
#include <hip/hip_runtime.h>

typedef __attribute__((ext_vector_type(2))) float v2f;
typedef __attribute__((ext_vector_type(8))) float v8f;
typedef __attribute__((ext_vector_type(4))) unsigned int v4u;
typedef __attribute__((ext_vector_type(8))) int v8i;
typedef __attribute__((ext_vector_type(4))) int v4i;

#define NROWS 4096
#define DIM   512
#define LDA   516            // A tile row stride (floats): 516 % 64 == 4 -> conflict-free
#define KCH   64             // K-chunk staged per TDM transfer
#define LDB   68             // B chunk row stride = 64 + 4 pad DWORDs (TDM pad_interval=5, pad_amount=3)
#define MARGIN 0.3f

// ---- dynamic LDS layout (floats) ----
// sA      : 16*LDA                      = 8256
// sB      : 8 waves * 2 bufs * 16*LDB   = 17408
// sRowSqn : 16
// sRowLab : 16 (ints)
// sHP/sHN : 128 + 128
// sCNT    : 128 (ints)
#define SA_OFF   0
#define SB_OFF   (16 * LDA)
#define SB_WAVE  (2 * 16 * LDB)
#define RSQ_OFF  (SB_OFF + 8 * SB_WAVE)
#define RLAB_OFF (RSQ_OFF + 16)
#define HP_OFF   (RLAB_OFF + 16)
#define HN_OFF   (HP_OFF + 128)
#define CNT_OFF  (HN_OFF + 128)
#define SMEM_DW  (CNT_OFF + 128)
#define SMEM_BYTES (SMEM_DW * 4)

// ---------------- kernel 1: row squared norms + zero the output scalar ----------------
__global__ void sqnorm_kernel(const float* __restrict__ x,
                              float* __restrict__ sqn,
                              float* __restrict__ out) {
    if (blockIdx.x == 0 && threadIdx.x == 0) out[0] = 0.0f;
    int gwave = (blockIdx.x * blockDim.x + threadIdx.x) >> 5;   // one wave32 per row
    int lane  = threadIdx.x & 31;
    if (gwave >= NROWS) return;
    const float* row = x + (size_t)gwave * DIM;
    float s = 0.0f;
    #pragma unroll 4
    for (int k = lane; k < DIM; k += 32) {
        float v = row[k];
        s = fmaf(v, v, s);
    }
    #pragma unroll
    for (int off = 16; off > 0; off >>= 1) s += __shfl_xor(s, off, 32);
    if (lane == 0) sqn[gwave] = s;
}

// ---- TDM: async-load one 16-row x 64-col f32 tile of X into LDS (padded rows) ----
__device__ __forceinline__ void tdm_load_tile(const float* gtile, unsigned lds_addr) {
    unsigned long long ga = (unsigned long long)(uintptr_t)gtile;
    // D# group 0: count=1, lds_addr, 57-bit global_addr, type=2 ("image")
    v4u g0 = { 1u,
               lds_addr,
               (unsigned)(ga & 0xFFFFFFFFu),
               (unsigned)((ga >> 32) & 0x1FFFFFFu) | (2u << 30) };
    // D# group 1:
    //  w0: workgroup_mask=0 | data_size=2(4B)<<16 | pad_enable<<20 | pad_interval=5<<22 | pad_amount=3<<25
    //  tensor_dim0=512 (bits 79:48), tensor_dim1=4096 (bits 111:80), tile_dim0=64 (127:112)
    //  tile_dim1=16 (143:128), tile_dim2=0, tensor_dim0_stride=512 (207:160)
    v8i g1 = { (int)((2u << 16) | (1u << 20) | (5u << 22) | (3u << 25)),
               (int)((512u & 0xFFFFu) << 16),
               (int)(((512u >> 16) & 0xFFFFu) | ((4096u & 0xFFFFu) << 16)),
               (int)(((4096u >> 16) & 0xFFFFu) | (64u << 16)),
               16,
               512,
               0,
               0 };
    v4i gz4 = { 0, 0, 0, 0 };                 // 2D tile: groups 2/3 unused
    v8i gz8 = { 0, 0, 0, 0, 0, 0, 0, 0 };     // 6-arg toolchain variant: extra group, zero-filled
    __builtin_amdgcn_tensor_load_to_lds(g0, g1, gz4, gz4, gz8, 0);
}

// ---------------- kernel 2: fused Gram (WMMA f32 16x16x4) + masked row reductions ----------------
__global__ __launch_bounds__(256, 1)
void triplet_kernel(const float* __restrict__ x,
                    const int*   __restrict__ labels,
                    const float* __restrict__ sqn,
                    float*       __restrict__ out) {
    extern __shared__ float smem[];
    float* sA      = smem + SA_OFF;
    float* sRowSqn = smem + RSQ_OFF;
    int*   sRowLab = (int*)(smem + RLAB_OFF);
    float* sHP     = smem + HP_OFF;     // [8][16]
    float* sHN     = smem + HN_OFF;     // [8][16]
    int*   sCNT    = (int*)(smem + CNT_OFF);

    const int tid  = threadIdx.x;
    const int wave = tid >> 5;
    const int lane = tid & 31;
    const int half = lane >> 4;          // K/N half select per WMMA layout
    const int l16  = lane & 15;
    const int rowbase = blockIdx.x * 16;

    float* sBw = smem + SB_OFF + wave * SB_WAVE;             // this wave's two B buffers
    const unsigned sBw_lds = (unsigned)(uintptr_t)sBw;        // LDS byte address

    // stage A tile (16 rows x 512) into LDS, coalesced
    for (int idx = tid; idx < 16 * DIM; idx += 256) {
        int r = idx >> 9;                // / 512
        int c = idx & (DIM - 1);
        sA[r * LDA + c] = x[(size_t)(rowbase + r) * DIM + c];
    }
    if (tid < 16) {
        sRowSqn[tid] = sqn[rowbase + tid];
        sRowLab[tid] = labels[rowbase + tid];
    }
    __syncthreads();

    // per-lane row-side constants: C-fragment reg r + lane-half -> row r + 8*half
    float rsq[8]; int rlab[8];
    #pragma unroll
    for (int r = 0; r < 8; ++r) {
        rsq[r]  = sRowSqn[r + 8 * half];
        rlab[r] = sRowLab[r + 8 * half];
    }

    float hp[8], hn[8]; int cnt[8];
    #pragma unroll
    for (int r = 0; r < 8; ++r) { hp[r] = -__builtin_inff(); hn[r] = __builtin_inff(); cnt[r] = 0; }

    const float* aLane = &sA[l16 * LDA + 2 * half];   // A fragment base for this lane

    // each wave handles every-8th 16-column tile; B staged by TDM, double-buffered
    for (int jt = wave; jt < NROWS / 16; jt += 8) {
        const int colbase = jt * 16;
        const int   clab = labels[colbase + l16];
        const float csq  = sqn[colbase + l16];
        const float* gcol = x + (size_t)colbase * DIM;   // tile rows = columns of Gram

        tdm_load_tile(gcol, sBw_lds);                    // chunk 0 -> buffer 0
        v8f c = {};

        #pragma unroll
        for (int ci = 0; ci < DIM / KCH; ++ci) {
            if (ci + 1 < DIM / KCH) {
                tdm_load_tile(gcol + (ci + 1) * KCH,
                              sBw_lds + (unsigned)(((ci + 1) & 1) * 16 * LDB * 4));
                __builtin_amdgcn_s_wait_tensorcnt(1);    // oldest (current chunk) complete
            } else {
                __builtin_amdgcn_s_wait_tensorcnt(0);
            }
            asm volatile("" ::: "memory");               // fence compiler across TDM arrival

            const float* aC = aLane + ci * KCH;
            const float* bC = sBw + (ci & 1) * 16 * LDB + l16 * LDB + 2 * half;
            #pragma unroll 8
            for (int k = 0; k < KCH; k += 4) {
                v2f a = *(const v2f*)(aC + k);           // LDS, conflict-free
                v2f b = *(const v2f*)(bC + k);           // LDS, conflict-free (TDM-padded)
                c = __builtin_amdgcn_wmma_f32_16x16x4_f32(
                        false, a, false, b, (short)0, c, false, false);
            }
        }

        // fused masked distance reductions; dist^2 = |r|^2 + |c|^2 - 2 r.c
        #pragma unroll
        for (int r = 0; r < 8; ++r) {
            float sq = fmaxf(rsq[r] + csq - 2.0f * c[r], 0.0f);
            float d  = sqrtf(sq);
            if (rlab[r] == clab) { hp[r] = fmaxf(hp[r], d); cnt[r] += 1; }
            else                 { hn[r] = fminf(hn[r], d); }
        }
    }

    // reduce across the 16 N-lanes within each half
    #pragma unroll
    for (int r = 0; r < 8; ++r) {
        #pragma unroll
        for (int off = 1; off < 16; off <<= 1) {
            hp[r]  = fmaxf(hp[r], __shfl_xor(hp[r], off, 32));
            hn[r]  = fminf(hn[r], __shfl_xor(hn[r], off, 32));
            cnt[r] +=             __shfl_xor(cnt[r], off, 32);
        }
    }
    if (l16 == 0) {
        #pragma unroll
        for (int r = 0; r < 8; ++r) {
            int row = r + 8 * half;
            sHP[wave * 16 + row]  = hp[r];
            sHN[wave * 16 + row]  = hn[r];
            sCNT[wave * 16 + row] = cnt[r];
        }
    }
    __syncthreads();

    // combine the 8 waves, compute per-row hinge, block-sum, one atomic per block
    if (tid < 16) {
        float HP = -__builtin_inff(), HN = __builtin_inff(); int C = 0;
        #pragma unroll
        for (int w = 0; w < 8; ++w) {
            HP = fmaxf(HP, sHP[w * 16 + tid]);
            HN = fminf(HN, sHN[w * 16 + tid]);
            C += sCNT[w * 16 + tid];
        }
        float per = (C > 1) ? fmaxf(HP - HN + MARGIN, 0.0f) : 0.0f;
        #pragma unroll
        for (int off = 1; off < 16; off <<= 1) per += __shfl_xor(per, off, 32);
        if (tid == 0) atomicAdd(out, per * (1.0f / NROWS));
    }
}

extern "C" void kernel_launch(void* const* d_in, const int* in_sizes, int n_in,
                              void* d_out, int out_size, void* d_ws, size_t ws_size,
                              hipStream_t stream) {
    const float* x      = (const float*)d_in[0];
    const int*   labels = (const int*)d_in[1];
    float*       out    = (float*)d_out;
    float*       sqn    = (float*)d_ws;          // 4096 floats of scratch

    // 4096 waves (one per row) / 8 waves per block = 512 blocks
    sqnorm_kernel<<<NROWS / 8, 256, 0, stream>>>(x, sqn, out);
    // one block per 16-row tile; ~104KB dynamic LDS (A tile + 8x2 TDM B buffers)
    triplet_kernel<<<NROWS / 16, 256, SMEM_BYTES, stream>>>(x, labels, sqn, out);
}